// GCN_2_52158082842624
// MI455X (gfx1250) — compile-verified
//
#include <hip/hip_runtime.h>

typedef __attribute__((ext_vector_type(2))) float v2f;
typedef __attribute__((ext_vector_type(8))) float v8f;

#define HID 96
#define NGRAPH 64
#define NCLS 10
#define TPB 256

__device__ inline void atomicAddF(float* p, float v) {
  __hip_atomic_fetch_add(p, v, __ATOMIC_RELAXED, __HIP_MEMORY_SCOPE_AGENT);
}

// ---------------- utility ----------------
__global__ void k_zero(unsigned* __restrict__ p, int n) {
  int i = blockIdx.x * blockDim.x + threadIdx.x;
  if (i < n) p[i] = 0u;
}

// ---------------- CSR build (dst-sorted) ----------------
__global__ void k_hist(const int* __restrict__ dst, int* __restrict__ cnt,
                       int* __restrict__ pos, int E) {
  int e = blockIdx.x * blockDim.x + threadIdx.x;
  if (e < E) pos[e] = atomicAdd(&cnt[dst[e]], 1);
}

__global__ void k_dinv(const int* __restrict__ cnt, float* __restrict__ dinv, int n) {
  int i = blockIdx.x * blockDim.x + threadIdx.x;
  if (i < n) dinv[i] = rsqrtf((float)cnt[i] + 1.0f);  // +1 self loop
}

// block-wise inclusive scan (Hillis-Steele in LDS)
__global__ void k_scan_block(const int* __restrict__ in, int* __restrict__ out,
                             int* __restrict__ partials, int n) {
  __shared__ int s[TPB];
  int i = blockIdx.x * TPB + threadIdx.x;
  s[threadIdx.x] = (i < n) ? in[i] : 0;
  __syncthreads();
  for (int off = 1; off < TPB; off <<= 1) {
    int t = (threadIdx.x >= off) ? s[threadIdx.x - off] : 0;
    __syncthreads();
    s[threadIdx.x] += t;
    __syncthreads();
  }
  if (i < n) out[i] = s[threadIdx.x];
  if (threadIdx.x == TPB - 1) partials[blockIdx.x] = s[TPB - 1];
}

__global__ void k_scan_finish(const int* __restrict__ incl, const int* __restrict__ pscan,
                              int* __restrict__ rowptr, int n) {
  int i = blockIdx.x * blockDim.x + threadIdx.x;
  if (i < n) {
    int b = i >> 8;
    int off = (b > 0) ? pscan[b - 1] : 0;
    rowptr[i + 1] = incl[i] + off;
  }
  if (i == 0) rowptr[0] = 0;
}

__global__ void k_scatter(const int* __restrict__ src, const int* __restrict__ dst,
                          const int* __restrict__ rowptr, const int* __restrict__ pos,
                          int* __restrict__ csr_src, int E) {
  int e = blockIdx.x * blockDim.x + threadIdx.x;
  if (e < E) csr_src[rowptr[dst[e]] + pos[e]] = src[e];
}

// ---------------- fp32 WMMA GEMM: C[M x 96] = A[M x K] @ B[K x 96] ----------
// one wave computes a full 16(M) x 96(N) strip via six 16x16 accumulators
__global__ void k_gemm96(const float* __restrict__ A, const float* __restrict__ B,
                         float* __restrict__ C, int M, int K) {
  int wave = (blockIdx.x * blockDim.x + threadIdx.x) >> 5;
  int lane = threadIdx.x & 31;
  if (wave >= (M >> 4)) return;          // wave-uniform: EXEC stays all-ones
  int half = lane >> 4;                  // 0 or 1
  int l15  = lane & 15;
  int kb   = half << 1;                  // K sub-offset 0 or 2
  const float* Arow = A + (size_t)(wave * 16 + l15) * K;
  v8f acc[6] = {};
  for (int k = 0; k < K; k += 4) {
    v2f a;
    a.x = Arow[k + kb];
    a.y = Arow[k + kb + 1];
    const float* Brow0 = B + (size_t)(k + kb) * HID + l15;
    const float* Brow1 = Brow0 + HID;
#pragma unroll
    for (int t = 0; t < 6; ++t) {
      v2f b;
      b.x = Brow0[t * 16];
      b.y = Brow1[t * 16];
      acc[t] = __builtin_amdgcn_wmma_f32_16x16x4_f32(
          false, a, false, b, (short)0, acc[t], false, false);
    }
  }
  float* Cp = C + (size_t)(wave * 16 + (half << 3)) * HID + l15;
#pragma unroll
  for (int t = 0; t < 6; ++t)
#pragma unroll
    for (int r = 0; r < 8; ++r)
      Cp[(size_t)r * HID + t * 16] = acc[t][r];
}

// --------- fused gather-aggregate + self-loop + bias + ReLU (wave/node) -----
__global__ void k_aggregate(const int* __restrict__ csr_src, const int* __restrict__ rowptr,
                            const float* __restrict__ dinv, const float* __restrict__ hw,
                            const float* __restrict__ bias, float* __restrict__ hout,
                            int Nn) {
  int node = (blockIdx.x * blockDim.x + threadIdx.x) >> 5;
  int lane = threadIdx.x & 31;
  if (node >= Nn) return;
  float dd = dinv[node];
  const float* hs = hw + (size_t)node * HID;
  float selfn = dd * dd;
  float a0 = hs[lane] * selfn;
  float a1 = hs[lane + 32] * selfn;
  float a2 = hs[lane + 64] * selfn;
  int beg = rowptr[node], end = rowptr[node + 1];
  for (int j = beg; j < end; ++j) {
    int s = csr_src[j];
    float nrm = dinv[s] * dd;
    const float* h = hw + (size_t)s * HID;
    a0 += h[lane] * nrm;
    a1 += h[lane + 32] * nrm;
    a2 += h[lane + 64] * nrm;
  }
  float* o = hout + (size_t)node * HID;
  float v;
  v = a0 + bias[lane];      o[lane]      = v > 0.f ? v : 0.f;
  v = a1 + bias[lane + 32]; o[lane + 32] = v > 0.f ? v : 0.f;
  v = a2 + bias[lane + 64]; o[lane + 64] = v > 0.f ? v : 0.f;
}

// ---------------- pooling ----------------
__global__ void k_pool(const float* __restrict__ h, const int* __restrict__ batch,
                       float* __restrict__ gsum, float* __restrict__ gcnt, int total) {
  int i = blockIdx.x * blockDim.x + threadIdx.x;
  if (i >= total) return;
  int node = i / HID, c = i % HID;
  int g = batch[node];
  atomicAddF(&gsum[g * HID + c], h[i]);
  if (c == 0) atomicAddF(&gcnt[g], 1.0f);
}

__global__ void k_pooldiv(float* __restrict__ gsum, const float* __restrict__ gcnt) {
  int i = blockIdx.x * blockDim.x + threadIdx.x;
  if (i >= NGRAPH * HID) return;
  gsum[i] /= fmaxf(gcnt[i / HID], 1.0f);
}

// ---------------- MLP head ----------------
__global__ void k_mlp1(const float* __restrict__ pooled, const float* __restrict__ lw1,
                       const float* __restrict__ lb1, float* __restrict__ z1) {
  int i = blockIdx.x * blockDim.x + threadIdx.x;
  if (i >= NGRAPH * HID) return;
  int g = i / HID, j = i % HID;
  float acc = lb1[j];
  const float* pg = pooled + g * HID;
  for (int k = 0; k < HID; ++k) acc += pg[k] * lw1[k * HID + j];
  z1[i] = acc > 0.f ? acc : 0.f;
}

__global__ void k_mlp2(const float* __restrict__ z1, const float* __restrict__ lw2,
                       const float* __restrict__ lb2, float* __restrict__ out) {
  int i = blockIdx.x * blockDim.x + threadIdx.x;
  if (i >= NGRAPH * NCLS) return;
  int g = i / NCLS, j = i % NCLS;
  float acc = lb2[j];
  const float* zg = z1 + g * HID;
  for (int k = 0; k < HID; ++k) acc += zg[k] * lw2[k * NCLS + j];
  out[i] = acc;
}

// ---------------- host launcher ----------------
static inline size_t alignUp(size_t x) { return (x + 255) & ~(size_t)255; }
static inline int cdiv(long long a, long long b) { return (int)((a + b - 1) / b); }

extern "C" void kernel_launch(void* const* d_in, const int* in_sizes, int n_in,
                              void* d_out, int out_size, void* d_ws, size_t ws_size,
                              hipStream_t stream) {
  const float* x    = (const float*)d_in[0];
  const int*   ei   = (const int*)d_in[1];   // [2, E]
  const int*   batch= (const int*)d_in[2];
  const float* W1   = (const float*)d_in[3];
  const float* b1   = (const float*)d_in[4];
  const float* W2   = (const float*)d_in[5];
  const float* b2   = (const float*)d_in[6];
  const float* W3   = (const float*)d_in[7];
  const float* b3   = (const float*)d_in[8];
  const float* lw1  = (const float*)d_in[9];
  const float* lb1  = (const float*)d_in[10];
  const float* lw2  = (const float*)d_in[11];
  const float* lb2  = (const float*)d_in[12];
  float* out = (float*)d_out;

  const int Nn = in_sizes[2];            // 50000 nodes (multiple of 16)
  const int E  = in_sizes[1] / 2;        // 800000 edges
  const int D  = in_sizes[0] / Nn;       // 128 input features
  const int* esrc = ei;
  const int* edst = ei + E;

  // workspace carve-out
  char* w = (char*)d_ws;
  float* dinv    = (float*)w; w += alignUp((size_t)Nn * 4);
  float* bufA    = (float*)w; w += alignUp((size_t)Nn * HID * 4);
  float* bufB    = (float*)w; w += alignUp((size_t)Nn * HID * 4);
  int*   cnt     = (int*)w;   w += alignUp((size_t)Nn * 4);
  int*   incl    = (int*)w;   w += alignUp((size_t)Nn * 4);
  int*   rowptr  = (int*)w;   w += alignUp((size_t)(Nn + 1) * 4);
  int*   pos     = (int*)w;   w += alignUp((size_t)E * 4);
  int*   csr_src = (int*)w;   w += alignUp((size_t)E * 4);
  int*   partials= (int*)w;   w += alignUp((size_t)TPB * 4);
  int*   pscan   = (int*)w;   w += alignUp((size_t)TPB * 4);
  int*   pdummy  = (int*)w;   w += alignUp((size_t)TPB * 4);
  float* gsum    = (float*)w; w += alignUp((size_t)(NGRAPH * HID + NGRAPH) * 4);
  float* gcnt    = gsum + NGRAPH * HID;
  float* z1      = (float*)w; w += alignUp((size_t)NGRAPH * HID * 4);

  const int nbScan = cdiv(Nn, TPB);      // 196 (<= 256, one-level re-scan ok)

  // ---- degree + CSR build (dst-sorted, stores src directly) ----
  k_zero<<<cdiv(Nn, TPB), TPB, 0, stream>>>((unsigned*)cnt, Nn);
  k_hist<<<cdiv(E, TPB), TPB, 0, stream>>>(edst, cnt, pos, E);
  k_dinv<<<cdiv(Nn, TPB), TPB, 0, stream>>>(cnt, dinv, Nn);
  k_scan_block<<<nbScan, TPB, 0, stream>>>(cnt, incl, partials, Nn);
  k_scan_block<<<1, TPB, 0, stream>>>(partials, pscan, pdummy, nbScan);
  k_scan_finish<<<cdiv(Nn, TPB), TPB, 0, stream>>>(incl, pscan, rowptr, Nn);
  k_scatter<<<cdiv(E, TPB), TPB, 0, stream>>>(esrc, edst, rowptr, pos, csr_src, E);

  const int gemmBlocks = cdiv((long long)(Nn / 16) * 32, TPB);
  const int aggBlocks  = cdiv((long long)Nn * 32, TPB);

  // ---- layer 1: X @ W1 -> gather-agg -> relu ----
  k_gemm96<<<gemmBlocks, TPB, 0, stream>>>(x, W1, bufA, Nn, D);
  k_aggregate<<<aggBlocks, TPB, 0, stream>>>(csr_src, rowptr, dinv, bufA, b1, bufB, Nn);
  // ---- layer 2 ----
  k_gemm96<<<gemmBlocks, TPB, 0, stream>>>(bufB, W2, bufA, Nn, HID);
  k_aggregate<<<aggBlocks, TPB, 0, stream>>>(csr_src, rowptr, dinv, bufA, b2, bufB, Nn);
  // ---- layer 3 ----
  k_gemm96<<<gemmBlocks, TPB, 0, stream>>>(bufB, W3, bufA, Nn, HID);
  k_aggregate<<<aggBlocks, TPB, 0, stream>>>(csr_src, rowptr, dinv, bufA, b3, bufB, Nn);

  // ---- global mean pool ----
  k_zero<<<cdiv(NGRAPH * HID + NGRAPH, TPB), TPB, 0, stream>>>((unsigned*)gsum,
                                                               NGRAPH * HID + NGRAPH);
  k_pool<<<cdiv((long long)Nn * HID, TPB), TPB, 0, stream>>>(bufB, batch, gsum, gcnt,
                                                             Nn * HID);
  k_pooldiv<<<cdiv(NGRAPH * HID, TPB), TPB, 0, stream>>>(gsum, gcnt);

  // ---- MLP head ----
  k_mlp1<<<cdiv(NGRAPH * HID, TPB), TPB, 0, stream>>>(gsum, lw1, lb1, z1);
  k_mlp2<<<cdiv(NGRAPH * NCLS, TPB), TPB, 0, stream>>>(z1, lw2, lb2, out);

  (void)n_in; (void)out_size; (void)ws_size;
}